// LReLu_3272765079902
// MI455X (gfx1250) — compile-verified
//
#include <hip/hip_runtime.h>
#include <hip/hip_bf16.h>
#include <stdint.h>

// ---------------------------------------------------------------------------
// filtered_lrelu (up=2, down=2, 12-tap FIR) as chained banded WMMA GEMMs.
//   y = D * act( U * X * U^T + bias ) * D^T      per (n,c) 64x64 slice
// U: 138x64 up operator (6 nnz/row), D: 64x138 down operator (12 nnz/row),
// padded to 144 for 16x16 tiling. Intermediates live entirely in LDS.
// All tile indices / K bounds are forced scalar (readfirstlane) so loop
// control is SALU and EXEC is provably all-ones at every v_wmma.
// ---------------------------------------------------------------------------

typedef float v2f  __attribute__((ext_vector_type(2)));
typedef float v8f  __attribute__((ext_vector_type(8)));
typedef unsigned int u32x4 __attribute__((ext_vector_type(4)));
typedef int   i32x4 __attribute__((ext_vector_type(4)));
typedef int   i32x8 __attribute__((ext_vector_type(8)));

#define IN   64    // input/output spatial size
#define UPN  138   // up-domain size: (64*2-1) + 11 + 11 - 12 + 1
#define UPP  144   // padded to multiple of 16
#define TAPS 12

// ---- operator builder: U[144][64], UT[64][144], D[64][144], DT[144][64] ----
__global__ void build_ops_kernel(const float* __restrict__ up,
                                 const float* __restrict__ down,
                                 float* __restrict__ ws) {
  float* U  = ws;                    // [UPP][IN]
  float* UT = ws + UPP * IN;         // [IN][UPP]
  float* D  = ws + 2 * UPP * IN;     // [IN][UPP]
  float* DT = ws + 3 * UPP * IN;     // [UPP][IN]
  int tid    = blockIdx.x * blockDim.x + threadIdx.x;
  int stride = blockDim.x * gridDim.x;
  for (int idx = tid; idx < UPP * IN; idx += stride) {
    int u = idx / IN;
    int i = idx - u * IN;            // i = input row (up) / output row w (down)
    int t = u - 2 * i;
    bool band = (u < UPN) && (t >= 0) && (t < TAPS);
    float uv = band ? 2.0f * up[t] : 0.0f;         // fu = up_filter * UP
    float dv = band ? down[TAPS - 1 - t] : 0.0f;   // flipped down filter
    U [u * IN  + i] = uv;
    UT[i * UPP + u] = uv;
    D [i * UPP + u] = dv;
    DT[u * IN  + i] = dv;
  }
}

__device__ __forceinline__ int sread(int v) {
  return __builtin_amdgcn_readfirstlane(v);
}

// ---- WMMA fragment helpers (32-bit 16x4 A / 4x16 B / 16x16 C layouts) ----
__device__ __forceinline__ v2f ldA(const float* p, int ld, int m0, int k0,
                                   int r, int h) {
  // lanes 0-15: K = 0,1 ; lanes 16-31: K = 2,3  (row M = m0 + r)
  const float* q = p + (m0 + r) * ld + k0 + 2 * h;
  v2f a; a.x = q[0]; a.y = q[1];
  return a;
}
__device__ __forceinline__ v2f ldB(const float* p, int ld, int k0, int n0,
                                   int r, int h) {
  // lanes 0-15: rows K = 0,1 ; lanes 16-31: rows K = 2,3 (col N = n0 + r)
  const float* q = p + (k0 + 2 * h) * ld + n0 + r;
  v2f b; b.x = q[0]; b.y = q[ld];
  return b;
}
__device__ __forceinline__ v8f wmma4(v2f a, v2f b, v8f c) {
  return __builtin_amdgcn_wmma_f32_16x16x4_f32(false, a, false, b,
                                               (short)0, c, false, false);
}

__launch_bounds__(256)
__global__ void filtered_lrelu_kernel(const float* __restrict__ x,
                                      const float* __restrict__ bias,
                                      const float* __restrict__ ws,
                                      float* __restrict__ out) {
  __shared__ float Xs[IN * IN];       //  16 KB  input slice
  __shared__ float Y2[UPP * UPP];     //  81 KB  up-domain (activated)
  __shared__ float Y13[UPP * IN];     //  36 KB  Y1 (stages 1-2) / Y3 (3-4)

  const float* U  = ws;
  const float* UT = ws + UPP * IN;
  const float* D  = ws + 2 * UPP * IN;
  const float* DT = ws + 3 * UPP * IN;

  const int slice = blockIdx.x;                 // n*128 + c
  const float bv  = bias[slice & 127];
  const float* xs = x   + (size_t)slice * (IN * IN);
  float*       os = out + (size_t)slice * (IN * IN);

  const int lane = threadIdx.x & 31;
  const int wid  = sread(threadIdx.x >> 5);     // scalar wave id (8 waves)
  const int r    = lane & 15;
  const int h    = lane >> 4;

  // ---- TDM: async DMA the 64x64 fp32 slice into LDS (wave 0 issues) ----
  if (wid == 0) {
    uint64_t ga = (uint64_t)(uintptr_t)xs;
    uint32_t la = (uint32_t)(uintptr_t)(&Xs[0]);  // low 32 bits = LDS byte addr
    u32x4 g0;
    g0[0] = 1u;                                   // count=1, user descriptor
    g0[1] = la;                                   // lds_addr
    g0[2] = (uint32_t)ga;                         // global_addr[31:0]
    g0[3] = (uint32_t)((ga >> 32) & 0x01FFFFFFu)  // global_addr[56:32]
          | (2u << 30);                           // type=2 ("image")
    i32x8 g1;
    g1[0] = (int)(2u << 16);        // workgroup_mask=0, data_size=4B
    g1[1] = (int)(64u << 16);       // tensor_dim0 = 64 (bits 79:48)
    g1[2] = (int)(64u << 16);       // tensor_dim1 = 64 (bits 111:80)
    g1[3] = (int)(64u << 16);       // tile_dim0   = 64 (bits 127:112)
    g1[4] = 64;                     // tile_dim1 = 64, tile_dim2 = 0
    g1[5] = 64;                     // tensor_dim0_stride = 64
    g1[6] = 0;
    g1[7] = 0;
    i32x4 z4 = {0, 0, 0, 0};
#if defined(__clang_major__) && (__clang_major__ >= 23)
    i32x8 z8 = {0, 0, 0, 0, 0, 0, 0, 0};
    __builtin_amdgcn_tensor_load_to_lds(g0, g1, z4, z4, z8, 0);
#else
    __builtin_amdgcn_tensor_load_to_lds(g0, g1, z4, z4, 0);
#endif
    __builtin_amdgcn_s_wait_tensorcnt(0);
  }
  __syncthreads();

  // ---- Stage 1: Y1(144x64) = U(144x64) * Xs(64x64), banded K over i ----
  for (int t = wid; t < 9 * 4; t += 8) {
    const int m0 = sread((t >> 2) * 16);
    const int n0 = sread((t & 3) * 16);
    int iLo = (m0 - 10) >> 1; if (iLo < 0)  iLo = 0;
    int iHi = (m0 + 15) >> 1; if (iHi > 63) iHi = 63;
    const int kLo = sread(iLo & ~3);
    const int kHi = sread(iHi);
    v8f acc = {};
    for (int k0 = kLo; k0 <= kHi; k0 += 4)
      acc = wmma4(ldA(U, IN, m0, k0, r, h), ldB(Xs, IN, k0, n0, r, h), acc);
#pragma unroll
    for (int j = 0; j < 8; ++j)
      Y13[(m0 + j + 8 * h) * IN + n0 + r] = acc[j];
  }
  __syncthreads();

  // ---- Stage 2: Y2 = Y1 * UT, fused bias + lrelu + gain, banded K ----
  const float GAIN = 1.41421356237309515f;
  for (int t = wid; t < 9 * 9; t += 8) {
    const int m0 = sread((t / 9) * 16);
    const int n0 = sread((t % 9) * 16);
    int iLo = (n0 - 10) >> 1; if (iLo < 0)  iLo = 0;
    int iHi = (n0 + 15) >> 1; if (iHi > 63) iHi = 63;
    const int kLo = sread(iLo & ~3);
    const int kHi = sread(iHi);
    v8f acc = {};
    for (int k0 = kLo; k0 <= kHi; k0 += 4)
      acc = wmma4(ldA(Y13, IN, m0, k0, r, h), ldB(UT, UPP, k0, n0, r, h), acc);
#pragma unroll
    for (int j = 0; j < 8; ++j) {
      float v = acc[j] + bv;
      v = (v >= 0.0f ? v : 0.2f * v) * GAIN;
      Y2[(m0 + j + 8 * h) * UPP + n0 + r] = v;
    }
  }
  __syncthreads();

  // ---- Stage 3: Y3(64x144) = D(64x144) * Y2(144x144), banded K over u ----
  for (int t = wid; t < 4 * 9; t += 8) {
    const int m0 = sread((t / 9) * 16);
    const int n0 = sread((t % 9) * 16);
    const int kLo = sread(2 * m0);                       // 4-aligned (m0=16t)
    int uHi = 2 * m0 + 41; if (uHi > UPN - 1) uHi = UPN - 1;
    const int kHi = sread(uHi);
    v8f acc = {};
    for (int k0 = kLo; k0 <= kHi; k0 += 4)
      acc = wmma4(ldA(D, UPP, m0, k0, r, h), ldB(Y2, UPP, k0, n0, r, h), acc);
#pragma unroll
    for (int j = 0; j < 8; ++j)
      Y13[(m0 + j + 8 * h) * UPP + n0 + r] = acc[j];   // now holds Y3[64][144]
  }
  __syncthreads();

  // ---- Stage 4: out(64x64) = Y3 * DT, banded K over u, store global ----
  for (int t = wid; t < 4 * 4; t += 8) {
    const int m0 = sread((t >> 2) * 16);
    const int n0 = sread((t & 3) * 16);
    const int kLo = sread(2 * n0);
    int uHi = 2 * n0 + 41; if (uHi > UPN - 1) uHi = UPN - 1;
    const int kHi = sread(uHi);
    v8f acc = {};
    for (int k0 = kLo; k0 <= kHi; k0 += 4)
      acc = wmma4(ldA(Y13, UPP, m0, k0, r, h), ldB(DT, IN, k0, n0, r, h), acc);
#pragma unroll
    for (int j = 0; j < 8; ++j)
      os[(m0 + j + 8 * h) * IN + n0 + r] = acc[j];
  }
}

extern "C" void kernel_launch(void* const* d_in, const int* in_sizes, int n_in,
                              void* d_out, int out_size, void* d_ws, size_t ws_size,
                              hipStream_t stream) {
  const float* x    = (const float*)d_in[0];
  const float* bias = (const float*)d_in[1];
  const float* up   = (const float*)d_in[2];
  const float* down = (const float*)d_in[3];
  float* ws = (float*)d_ws;          // needs 4 * 144 * 64 * 4 = 147456 bytes

  build_ops_kernel<<<36, 256, 0, stream>>>(up, down, ws);          // 9216 elems
  filtered_lrelu_kernel<<<32 * 128, 256, 0, stream>>>(x, bias, ws,
                                                      (float*)d_out);
}